// Net_shadow_27977416966300
// MI455X (gfx1250) — compile-verified
//
#include <hip/hip_runtime.h>

#define N_NODES 100000
#define N_EDGES 3200000
#define F_IN    512
#define HID     16
#define C_OUT   40
#define C_PAD   48
#define NTILES  (N_NODES / 16)   // 6250, exact (100000 % 16 == 0)

typedef float v2f __attribute__((ext_vector_type(2)));
typedef float v8f __attribute__((ext_vector_type(8)));

// ---------------------------------------------------------------- utilities
__global__ void zero_kernel(float* __restrict__ p, int n) {
  int i = blockIdx.x * blockDim.x + threadIdx.x;
  if (i < n) p[i] = 0.0f;
}

__global__ void degree_kernel(const int* __restrict__ src, const int* __restrict__ dst,
                              unsigned* __restrict__ deg_out, unsigned* __restrict__ deg_in) {
  int e = blockIdx.x * blockDim.x + threadIdx.x;
  if (e < N_EDGES) {
    atomicAdd(&deg_out[src[e]], 1u);
    atomicAdd(&deg_in[dst[e]], 1u);
  }
}

__global__ void norm_kernel(const unsigned* __restrict__ deg_out, const unsigned* __restrict__ deg_in,
                            float* __restrict__ norm_src, float* __restrict__ norm_dst) {
  int i = blockIdx.x * blockDim.x + threadIdx.x;
  if (i < N_NODES) {
    norm_src[i] = rsqrtf(fmaxf((float)deg_out[i], 1.0f));
    norm_dst[i] = rsqrtf(fmaxf((float)deg_in[i], 1.0f));
  }
}

// ---------------------------------------------------------------- GEMM 1
// h1[N,16] = (X * norm_src[:,None]) @ W1      (full fp32 via V_WMMA_F32_16X16X4_F32)
// One wave per 16-row tile. A: lanes 0-15 carry K=kbase+{0,1}, lanes 16-31 K=kbase+{2,3}.
// B: lane n (mod 16) carries column n; same K split across lane halves.
__global__ __launch_bounds__(256) void gemm1_wmma(const float* __restrict__ X,
                                                  const float* __restrict__ norm_src,
                                                  const float* __restrict__ W1,
                                                  float* __restrict__ h1) {
  __shared__ float w1s[F_IN * HID];                       // 32 KB of the 320 KB WGP LDS
  for (int i = threadIdx.x; i < F_IN * HID; i += blockDim.x) w1s[i] = W1[i];
  __syncthreads();

  int wave = threadIdx.x >> 5;                            // wave32
  int lane = threadIdx.x & 31;
  int tile = blockIdx.x * 8 + wave;
  if (tile >= NTILES) return;                             // wave-uniform guard

  int m    = lane & 15;
  int koff = (lane >> 4) << 1;                            // 0 or 2
  int row  = tile * 16 + m;
  float scale = norm_src[row];
  const float* xrow = X + (long)row * F_IN;

  v8f acc = {};
  for (int k = 0; k < F_IN; k += 4) {
    v2f a = *(const v2f*)(xrow + k + koff);               // global_load_b64, 8B aligned
    a *= scale;
    v2f b;
    b.x = w1s[(k + koff) * HID + m];
    b.y = w1s[(k + koff + 1) * HID + m];
    acc = __builtin_amdgcn_wmma_f32_16x16x4_f32(false, a, false, b,
                                                (short)0, acc, false, false);
  }
  // C/D layout: VGPR j -> M=j (lanes 0-15) / M=j+8 (lanes 16-31), N = lane&15
  int mbase = tile * 16 + ((lane >> 4) << 3);
#pragma unroll
  for (int j = 0; j < 8; ++j)
    h1[(mbase + j) * HID + m] = acc[j];
}

// ---------------------------------------------------------------- scatter layer 1
__global__ void scatter16(const int* __restrict__ src, const int* __restrict__ dst,
                          const float* __restrict__ h1, float* __restrict__ agg1) {
  int t = blockIdx.x * blockDim.x + threadIdx.x;
  if (t >= N_EDGES * 4) return;
  int e = t >> 2, q = t & 3;
  int s = src[e], d = dst[e];
  float4 v = *(const float4*)(h1 + s * HID + q * 4);      // L2-resident gather
  float* base = agg1 + d * HID + q * 4;
  atomicAdd(base + 0, v.x);
  atomicAdd(base + 1, v.y);
  atomicAdd(base + 2, v.z);
  atomicAdd(base + 3, v.w);
}

__global__ void relu_bias_kernel(float* __restrict__ agg1, const float* __restrict__ norm_dst,
                                 const float* __restrict__ b1) {
  int t = blockIdx.x * blockDim.x + threadIdx.x;
  if (t < N_NODES * HID) {
    int i = t >> 4, f = t & 15;
    agg1[t] = fmaxf(agg1[t] * norm_dst[i] + b1[f], 0.0f); // in place -> x2
  }
}

// ---------------------------------------------------------------- GEMM 2
// h2[N,40] = x2[N,16] @ W2[16,40], W2 zero-padded to [16,48] (3 N-tiles of 16)
__global__ __launch_bounds__(256) void gemm2_wmma(const float* __restrict__ x2,
                                                  const float* __restrict__ W2,
                                                  float* __restrict__ h2) {
  __shared__ float w2s[HID * C_PAD];                      // 3 KB
  for (int i = threadIdx.x; i < HID * C_PAD; i += blockDim.x) {
    int k = i / C_PAD, c = i - k * C_PAD;
    w2s[i] = (c < C_OUT) ? W2[k * C_OUT + c] : 0.0f;
  }
  __syncthreads();

  int wave = threadIdx.x >> 5;
  int lane = threadIdx.x & 31;
  int tile = blockIdx.x * 8 + wave;
  if (tile >= NTILES) return;

  int m    = lane & 15;
  int koff = (lane >> 4) << 1;
  int row  = tile * 16 + m;
  const float* xrow = x2 + row * HID;
  int mbase = tile * 16 + ((lane >> 4) << 3);

#pragma unroll
  for (int c = 0; c < 3; ++c) {
    v8f acc = {};
#pragma unroll
    for (int k = 0; k < HID; k += 4) {
      v2f a = *(const v2f*)(xrow + k + koff);
      v2f b;
      b.x = w2s[(k + koff) * C_PAD + c * 16 + m];
      b.y = w2s[(k + koff + 1) * C_PAD + c * 16 + m];
      acc = __builtin_amdgcn_wmma_f32_16x16x4_f32(false, a, false, b,
                                                  (short)0, acc, false, false);
    }
    int col = c * 16 + m;
    if (col < C_OUT) {                                    // divergence only at store
#pragma unroll
      for (int j = 0; j < 8; ++j)
        h2[(mbase + j) * C_OUT + col] = acc[j];
    }
  }
}

// ---------------------------------------------------------------- scatter layer 2 (into d_out)
__global__ void scatter40(const int* __restrict__ src, const int* __restrict__ dst,
                          const float* __restrict__ h2, float* __restrict__ out) {
  int t = blockIdx.x * blockDim.x + threadIdx.x;
  if (t >= N_EDGES * 10) return;
  int e = t / 10, q = t - e * 10;
  int s = src[e], d = dst[e];
  float4 v = *(const float4*)(h2 + s * C_OUT + q * 4);
  float* base = out + d * C_OUT + q * 4;
  atomicAdd(base + 0, v.x);
  atomicAdd(base + 1, v.y);
  atomicAdd(base + 2, v.z);
  atomicAdd(base + 3, v.w);
}

__global__ void finalize_kernel(float* __restrict__ out, const float* __restrict__ norm_dst,
                                const float* __restrict__ b2) {
  int t = blockIdx.x * blockDim.x + threadIdx.x;
  if (t < N_NODES * C_OUT) {
    int i = t / C_OUT, c = t - i * C_OUT;
    out[t] = out[t] * norm_dst[i] + b2[c];
  }
}

// ---------------------------------------------------------------- launch
extern "C" void kernel_launch(void* const* d_in, const int* in_sizes, int n_in,
                              void* d_out, int out_size, void* d_ws, size_t ws_size,
                              hipStream_t stream) {
  const float* X   = (const float*)d_in[0];   // [N, 512]
  const float* W1  = (const float*)d_in[1];   // [512, 16]
  const float* b1  = (const float*)d_in[2];   // [16]
  const float* W2  = (const float*)d_in[3];   // [16, 40]
  const float* b2  = (const float*)d_in[4];   // [40]
  const int*   src = (const int*)d_in[5];     // [E]
  const int*   dst = (const int*)d_in[6];     // [E]
  float* out = (float*)d_out;                 // [N, 40]

  // workspace layout (floats): deg_out N | deg_in N | agg1 16N | norm_src N |
  //                            norm_dst N | h1 16N | h2 40N   => 76N floats (~30.4 MB)
  float* ws = (float*)d_ws;
  unsigned* deg_out  = (unsigned*)ws;          // N
  unsigned* deg_in   = (unsigned*)(ws + N_NODES);
  float* agg1        = ws + 2  * N_NODES;      // 16N (becomes x2 in place)
  float* norm_src    = ws + 18 * N_NODES;
  float* norm_dst    = ws + 19 * N_NODES;
  float* h1          = ws + 20 * N_NODES;      // 16N
  float* h2          = ws + 36 * N_NODES;      // 40N

  const int B = 256;
  // re-zero all accumulators every call (graph replay determinism)
  int nz = 18 * N_NODES;                                   // deg_out, deg_in, agg1
  zero_kernel<<<(nz + B - 1) / B, B, 0, stream>>>(ws, nz);
  int no = N_NODES * C_OUT;
  zero_kernel<<<(no + B - 1) / B, B, 0, stream>>>(out, no);

  degree_kernel<<<(N_EDGES + B - 1) / B, B, 0, stream>>>(src, dst, deg_out, deg_in);
  norm_kernel<<<(N_NODES + B - 1) / B, B, 0, stream>>>(deg_out, deg_in, norm_src, norm_dst);

  int gemm_blocks = (NTILES + 7) / 8;                      // 8 waves (tiles) per block
  gemm1_wmma<<<gemm_blocks, B, 0, stream>>>(X, norm_src, W1, h1);

  int s16 = N_EDGES * 4;
  scatter16<<<(s16 + B - 1) / B, B, 0, stream>>>(src, dst, h1, agg1);
  relu_bias_kernel<<<(N_NODES * HID + B - 1) / B, B, 0, stream>>>(agg1, norm_dst, b1);

  gemm2_wmma<<<gemm_blocks, B, 0, stream>>>(agg1, W2, h2);

  int s40 = N_EDGES * 10;
  scatter40<<<(s40 + B - 1) / B, B, 0, stream>>>(src, dst, h2, out);
  finalize_kernel<<<(no + B - 1) / B, B, 0, stream>>>(out, norm_dst, b2);
}